// GaussianConditional_82325933130458
// MI455X (gfx1250) — compile-verified
//
#include <hip/hip_runtime.h>

typedef __attribute__((ext_vector_type(2))) float v2f;
typedef __attribute__((ext_vector_type(4))) float v4f;
typedef __attribute__((ext_vector_type(8))) float v8f;

#define TCHUNK 2          // t-values per wave
#define WPB    8          // waves per block (256 threads)

// T=2048, N=64, D=1024
__global__ __launch_bounds__(256) void gnll_wmma_kernel(
    const float* __restrict__ y,     // (T,N)
    const float* __restrict__ x,     // (T,N,D)
    const float* __restrict__ W,     // (2,D)
    const int*   __restrict__ lens,  // (N)
    float* __restrict__ out)         // (N)
{
    constexpr int N = 64, D = 1024;
    constexpr float HALF_LOG_2PI = 0.91893853320467274178f;

    const int tid  = threadIdx.x;
    const int lane = tid & 31;
    const int w    = tid >> 5;
    const int col  = lane & 15;      // A: row within 16-row tile; B/C: column
    const int h    = lane >> 4;      // lane-half selects K {0,1} vs K {2,3}

    const int nb = blockIdx.x & 3;   // 4 groups of 16 sequences
    const int tb = blockIdx.x >> 2;  // 128 t-blocks
    const int n0 = nb * 16;

    // [0..2047] = LDS copy of W (2 rows x 1024); [2048..3071] = zero pad for
    // the 14 inactive B columns (their per-k stream walks through the pad).
    __shared__ __align__(16) float wlds[3 * 1024];
    __shared__ __align__(16) float red[WPB][2][16];
    __shared__ float bred[16];

    for (int idx = tid; idx < 2 * D; idx += 256) wlds[idx] = W[idx];
    for (int idx = tid; idx < D; idx += 256)     wlds[2 * D + idx] = 0.0f;
    if (tid < 16) bred[tid] = 0.0f;
    __syncthreads();

    // B fragment base in LDS: columns 0,1 carry W rows; others read zeros.
    const int bbase = (col < 2) ? (col * D + h * 8) : (2 * D);

    const int myn = n0 + col;          // this lane's sequence (as A-row / C-col)
    const int len = lens[myn];

    const int tbase = tb * (WPB * TCHUNK) + w * TCHUNK;
    float acc = 0.0f;

    for (int i = 0; i < TCHUNK; ++i) {
        const int t = tbase + i;
        const float* px = x + ((size_t)t * N + myn) * D + h * 8;

        if (i + 1 < TCHUNK) {
            // warm next t-slab (gfx1250 global_prefetch_b8)
            __builtin_prefetch((const void*)(px + (size_t)N * D), 0, 0);
        }

        v8f c0 = {};
        v8f c1 = {};

        // Logical-K permutation: low lanes carry k..k+3, high lanes k+8..k+11,
        // so one float4 per lane feeds two 16x16x4 WMMAs; B (from LDS) uses
        // the same pattern, keeping A/B element pairing consistent.
        #pragma unroll 2
        for (int k = 0; k < D; k += 16) {
            v4f a0 = *(const v4f*)(px + k);
            v4f a1 = *(const v4f*)(px + k + 4);
            v4f b0 = *(const v4f*)&wlds[bbase + k];
            v4f b1 = *(const v4f*)&wlds[bbase + k + 4];

            v2f A, B;
            A.x = a0.x; A.y = a0.y; B.x = b0.x; B.y = b0.y;   // K {k,k+1,k+8,k+9}
            c0 = __builtin_amdgcn_wmma_f32_16x16x4_f32(false, A, false, B,
                                                       (short)0, c0, false, false);
            A.x = a0.z; A.y = a0.w; B.x = b0.z; B.y = b0.w;   // K {k+2,k+3,k+10,k+11}
            c1 = __builtin_amdgcn_wmma_f32_16x16x4_f32(false, A, false, B,
                                                       (short)0, c1, false, false);
            A.x = a1.x; A.y = a1.y; B.x = b1.x; B.y = b1.y;   // K {k+4,k+5,k+12,k+13}
            c0 = __builtin_amdgcn_wmma_f32_16x16x4_f32(false, A, false, B,
                                                       (short)0, c0, false, false);
            A.x = a1.z; A.y = a1.w; B.x = b1.z; B.y = b1.w;   // K {k+6,k+7,k+14,k+15}
            c1 = __builtin_amdgcn_wmma_f32_16x16x4_f32(false, A, false, B,
                                                       (short)0, c1, false, false);
        }
        v8f c = c0 + c1;

        // C layout: vgpr j, lanes 0-15 -> M=j, lanes 16-31 -> M=8+j; N=lane%16.
        // Columns 0 (mu) and 1 (var-logit) live in lanes {0,16} and {1,17}.
        if (col < 2) {
            #pragma unroll
            for (int j = 0; j < 8; ++j)
                red[w][col][h * 8 + j] = c[j];
        }
        __syncthreads();

        if (lane < 16) {
            float mu = red[w][0][lane];
            float s  = red[w][1][lane];
            float var = 1.0f / (1.0f + __expf(-s));
            var = fmaxf(var, 1e-6f);
            float yv = y[(size_t)t * N + n0 + lane];
            float d  = yv - mu;
            float nll = 0.5f * (__logf(var) + d * d / var) + HALF_LOG_2PI;
            if (t < len) acc += nll;
        }
        __syncthreads();
    }

    // block-level reduction over the 8 waves (all share n0), then one global
    // atomic per sequence per block (~128 atomics per output address total).
    if (lane < 16) atomicAdd(&bred[lane], acc);
    __syncthreads();
    if (tid < 16) atomicAdd(&out[n0 + tid], bred[tid]);
}

__global__ void gnll_zero_out(float* __restrict__ out) {
    out[threadIdx.x] = 0.0f;
}

extern "C" void kernel_launch(void* const* d_in, const int* in_sizes, int n_in,
                              void* d_out, int out_size, void* d_ws, size_t ws_size,
                              hipStream_t stream) {
    (void)in_sizes; (void)n_in; (void)d_ws; (void)ws_size; (void)out_size;
    const float* y    = (const float*)d_in[0];
    const float* x    = (const float*)d_in[1];
    const float* W    = (const float*)d_in[2];
    const int*   lens = (const int*)d_in[3];
    float* out = (float*)d_out;

    gnll_zero_out<<<1, 64, 0, stream>>>(out);

    // 4 n-groups * (2048 / (8 waves * TCHUNK)) t-blocks = 512 blocks
    dim3 grid(4 * (2048 / (WPB * TCHUNK)));
    gnll_wmma_kernel<<<grid, 256, 0, stream>>>(y, x, W, lens, out);
}